// RoIAlign1D_19069654794274
// MI455X (gfx1250) — compile-verified
//
#include <hip/hip_runtime.h>
#include <cstdint>

// ---------------------------------------------------------------------------
// RoIAlign1D for MI455X (gfx1250, wave32)
//   feat      : [B=8, LV=4096, D=256] f32   (33.5 MB -> L2 resident)
//   boxes     : [N=2048, 2] f32 (y1,y2 sorted)
//   batch_idx : [N] i32
//   out       : [N, P=32, D=256] f32
//
// One wave per (box, bin). Row indices over samples are monotone with step
// <= 1, so a 4-slot LDS ring (slot = row & 3) staged with CDNA5
// GLOBAL_LOAD_ASYNC_TO_LDS_B128 fetches each feature row from L2 exactly
// once (~1 KB/sample instead of 2 KB), overlapped with the FMA of the
// current sample via ASYNCcnt double-buffering.
// ---------------------------------------------------------------------------

namespace {
constexpr int kB  = 8;
constexpr int kLV = 4096;
constexpr int kD  = 256;          // channels; 1 row = 1024 bytes
constexpr int kN  = 2048;
constexpr int kP  = 32;
constexpr int kWaves   = 8;       // waves per block
constexpr int kThreads = kWaves * 32;
}

// Generic -> LDS(addrspace 3) offset, valid as VDST for async-to-LDS ops
// (the VDST value plus the instruction offset addresses LDS, same base as DS instructions).
__device__ __forceinline__ uint32_t lds_byte_offset(const void* p) {
  return (uint32_t)(uintptr_t)(const __attribute__((address_space(3))) void*)p;
}

// Copy one 1 KB feature row into LDS with two async b128 ops (GVS mode:
// saddr = feat base SGPR pair, vaddr = 32-bit per-lane byte offset).
// IOFFSET is added to BOTH the LDS and global addresses per the ISA, so
// offset:512 covers the second half of the row with the same vaddr/vdst.
__device__ __forceinline__ void async_copy_row_1kb(uint32_t lds_byte,
                                                   uint32_t glob_byte,
                                                   const void* feat_base) {
  asm volatile(
      "global_load_async_to_lds_b128 %0, %1, %2\n\t"
      "global_load_async_to_lds_b128 %0, %1, %2 offset:512"
      :
      : "v"(lds_byte), "v"(glob_byte), "s"(feat_base)
      : "memory");
}

struct Samp {
  int   lo, hi;
  float wlo, whi;
};

// Exact port of the reference bilinear() weight/index math.
__device__ __forceinline__ Samp make_samp(float y) {
  const bool  valid = (y >= -1.0f) && (y <= (float)kLV);
  float       yc    = fmaxf(y, 0.0f);
  int         ylo   = (int)floorf(yc);
  const bool  hic   = (ylo >= kLV - 1);
  ylo               = hic ? (kLV - 1) : ylo;
  const int   yhi   = hic ? (kLV - 1) : (ylo + 1);
  yc                = hic ? (float)ylo : yc;
  const float ly    = yc - (float)ylo;
  Samp s;
  s.lo  = ylo;
  s.hi  = yhi;
  s.wlo = valid ? (1.0f - ly) : 0.0f;
  s.whi = valid ? ly : 0.0f;
  return s;
}

__global__ __launch_bounds__(kThreads) void roi_align_1d_kernel(
    const float* __restrict__ feat, const float* __restrict__ boxes,
    const int* __restrict__ batch_idx, float* __restrict__ out) {
  // Per-wave LDS: 4 ring slots x 1 row x 64 float4 (4 KB). 32 KB per block.
  __shared__ float4 smem[kWaves * 4 * 64];

  const int tid  = threadIdx.x;
  const int lane = tid & 31;
  const int wave = tid >> 5;
  const int job  = blockIdx.x * kWaves + wave;   // 0 .. N*P-1
  const int n    = job >> 5;                     // box index
  const int p    = job & 31;                     // bin index

  // Box parameters (wave-uniform; compiler scalarizes onto s_*_f32 ops).
  const float y1    = boxes[2 * n + 0];
  const float y2    = boxes[2 * n + 1];
  const float roi_h = y2 - y1;
  const float start = y1 - 0.5f;
  const float bin_h = roi_h * (1.0f / (float)kP);
  const int   gh    = (int)ceilf(bin_h);
  const int   ghc   = (gh < 1) ? 1 : gh;
  const float count = (float)ghc;
  const float sub   = bin_h / (float)ghc;
  const float off0  = start + (float)p * bin_h;
  const int   brow  = batch_idx[n] * kLV;        // row base in feat_flat

  float4* wbuf = &smem[wave * 256];              // this wave's 256 float4
  const uint32_t lds_base = lds_byte_offset(wbuf);
  const uint32_t lane16   = (uint32_t)lane * 16u;

  // acc0 -> channels [4*lane, 4*lane+4); acc1 -> channels [128+4*lane, ...)
  float4 acc0 = make_float4(0.f, 0.f, 0.f, 0.f);
  float4 acc1 = make_float4(0.f, 0.f, 0.f, 0.f);

  // Stage feature row r into ring slot (r & 3).
  auto stage_row = [&](int r) {
    async_copy_row_1kb(lds_base + (uint32_t)(r & 3) * 1024u + lane16,
                       (uint32_t)(brow + r) * 1024u + lane16, feat);
  };

  // Prologue: stage the first sample's row(s). top = highest staged row.
  int top = 0;
  if (gh > 0) {
    const Samp s0 = make_samp(off0 + 0.5f * sub);
    stage_row(s0.lo);
    if (s0.hi > s0.lo) stage_row(s0.hi);
    top = s0.hi;
  }

  for (int g = 0; g < gh; ++g) {
    const Samp cur = make_samp(off0 + ((float)g + 0.5f) * sub);

    // Stage at most one new row for the next sample (rows are monotone with
    // step <= 1, so nx.lo is always <= cur.hi and already resident).
    bool issued = false;
    if (g + 1 < gh) {
      const Samp nx = make_samp(off0 + ((float)g + 1.5f) * sub);
      if (nx.hi > top) {
        // Slot (nx.hi & 3) was last read >= 2 samples ago; retire those
        // ds_loads before the async write overlaps them.
        asm volatile("s_wait_dscnt 0x0" ::: "memory");
        stage_row(nx.hi);
        top = nx.hi;
        issued = true;
      }
    }
    // In-order ASYNCcnt retirement: everything except the (optional) row we
    // just issued must be complete before reading this sample's slots.
    if (issued) {
      asm volatile("s_wait_asynccnt 0x2" ::: "memory");
    } else {
      asm volatile("s_wait_asynccnt 0x0" ::: "memory");
    }

    const float4* rl = wbuf + (cur.lo & 3) * 64; // row y_lo
    const float4* rh = wbuf + (cur.hi & 3) * 64; // row y_hi
    const float4  a0 = rl[lane];                 // lo row, ch 4*lane..
    const float4  a1 = rl[lane + 32];            // lo row, ch 128+4*lane..
    const float4  b0 = rh[lane];                 // hi row, ch 4*lane..
    const float4  b1 = rh[lane + 32];            // hi row, ch 128+4*lane..

    const float wl = cur.wlo, wh = cur.whi;
    acc0.x = fmaf(wl, a0.x, fmaf(wh, b0.x, acc0.x));
    acc0.y = fmaf(wl, a0.y, fmaf(wh, b0.y, acc0.y));
    acc0.z = fmaf(wl, a0.z, fmaf(wh, b0.z, acc0.z));
    acc0.w = fmaf(wl, a0.w, fmaf(wh, b0.w, acc0.w));
    acc1.x = fmaf(wl, a1.x, fmaf(wh, b1.x, acc1.x));
    acc1.y = fmaf(wl, a1.y, fmaf(wh, b1.y, acc1.y));
    acc1.z = fmaf(wl, a1.z, fmaf(wh, b1.z, acc1.z));
    acc1.w = fmaf(wl, a1.w, fmaf(wh, b1.w, acc1.w));
  }

  const float inv = 1.0f / count;
  float4 o0 = make_float4(acc0.x * inv, acc0.y * inv, acc0.z * inv, acc0.w * inv);
  float4 o1 = make_float4(acc1.x * inv, acc1.y * inv, acc1.z * inv, acc1.w * inv);

  float4* orow = (float4*)(out + (size_t)job * kD);
  orow[lane]      = o0;
  orow[lane + 32] = o1;
}

extern "C" void kernel_launch(void* const* d_in, const int* in_sizes, int n_in,
                              void* d_out, int out_size, void* d_ws,
                              size_t ws_size, hipStream_t stream) {
  (void)in_sizes; (void)n_in; (void)out_size; (void)d_ws; (void)ws_size;
  const float* feat  = (const float*)d_in[0];
  const float* boxes = (const float*)d_in[1];
  const int*   bidx  = (const int*)d_in[2];
  float*       out   = (float*)d_out;

  const int blocks = (kN * kP) / kWaves;  // 8192 blocks x 256 threads
  roi_align_1d_kernel<<<blocks, kThreads, 0, stream>>>(feat, boxes, bidx, out);
}